// SparseInteractionAdditiveRiskModel_66022237274451
// MI455X (gfx1250) — compile-verified
//
#include <hip/hip_runtime.h>
#include <math.h>

typedef __attribute__((ext_vector_type(2))) float v2f;
typedef __attribute__((ext_vector_type(8))) float v8f;

#define BROWS   262144
#define NF      10          // features
#define HU      64
#define NP      9           // pairs
#define HP      64
#define NCOLS   1216        // 10*64 + 9*64 hidden units total
#define KP      12          // padded K: 10 features + bias(=1) + zero pad
#define NTILES  76          // 1216 / 16
#define THREADS 256
#define WAVES   8
#define TPW     4           // row-tiles per wave
#define RT_PER_BLOCK (WAVES * TPW)              // 32
#define NBLOCKS (BROWS / 16 / RT_PER_BLOCK)     // 512

// Branch-free exact-erf GELU (Abramowitz-Stegun 7.1.26, |err| <= 1.5e-7):
//   erf(|x|) = 1 - (a1 t + a2 t^2 + a3 t^3 + a4 t^4 + a5 t^5) e^{-x^2}
//   t = 1/(1 + p|x|)
// Uses v_rcp_f32 + v_exp_f32 (TRANS ops -> co-execute with VALU/WMMA),
// no exec-mask divergence.
__device__ __forceinline__ float gelu_exact(float v) {
    const float xs = v * 0.70710678118654752440f;   // x / sqrt(2)
    const float ax = __builtin_fabsf(xs);
    const float t  = __builtin_amdgcn_rcpf(__builtin_fmaf(0.3275911f, ax, 1.0f));
    float p;
    p = __builtin_fmaf(1.061405429f, t, -1.453152027f);
    p = __builtin_fmaf(p, t,  1.421413741f);
    p = __builtin_fmaf(p, t, -0.284496736f);
    p = __builtin_fmaf(p, t,  0.254829592f);
    p = p * t;
    const float e   = __expf(-ax * ax);             // v_exp_f32, branch-free
    const float era = __builtin_fmaf(-p, e, 1.0f);  // erf(|xs|)
    const float erf_xs = __builtin_copysignf(era, xs);
    return 0.5f * v * (1.0f + erf_xs);
}

__global__ __launch_bounds__(THREADS) void nam_wmma_kernel(
    const float* __restrict__ x,
    const float* __restrict__ raw_mean, const float* __restrict__ raw_std,
    const float* __restrict__ w1f, const float* __restrict__ b1f,
    const float* __restrict__ w2f, const float* __restrict__ b2f,
    const float* __restrict__ w1p, const float* __restrict__ b1p,
    const float* __restrict__ w2p, const float* __restrict__ b2p,
    const int* __restrict__ crp_index_p,
    float* __restrict__ out)
{
    // LDS: W1 transposed, [n][k] with k-stride 12 dwords (conflict-free:
    // 12*n mod 64 spans 16 distinct 4-dword groups), plus second-layer weights.
    __shared__ float sW1T[NCOLS * KP];   // 58368 B
    __shared__ float sW2[NCOLS];         //  4864 B

    const int tid = threadIdx.x;

    // ---- build dense first-layer matrix in LDS ----
    for (int i = tid; i < NCOLS * KP; i += THREADS) sW1T[i] = 0.0f;
    __syncthreads();
    for (int j = tid; j < NCOLS; j += THREADS) {
        if (j < NF * HU) {                       // FeatureNet column
            int f = j >> 6, h = j & 63;
            sW1T[j * KP + f]  = w1f[f * HU + h];
            sW1T[j * KP + 10] = b1f[f * HU + h];
            sW2[j]            = w2f[f * HU + h];
        } else {                                 // PairNet column: (age, feat p+1)
            int q = j - NF * HU;
            int p = q >> 6, h = q & 63;
            sW1T[j * KP + 0]     = w1p[(p * 2 + 0) * HP + h];
            sW1T[j * KP + p + 1] = w1p[(p * 2 + 1) * HP + h];
            sW1T[j * KP + 10]    = b1p[p * HP + h];
            sW2[j]               = w2p[p * HP + h];
        }
    }
    __syncthreads();

    // constant output bias: sum of all second-layer biases
    float cbias = 0.0f;
    for (int f = 0; f < NF; ++f) cbias += b2f[f];
    for (int p = 0; p < NP; ++p) cbias += b2p[p];

    const int crp  = *crp_index_p;
    const int lane = tid & 31;
    const int wave = tid >> 5;
    const int half = lane >> 4;     // 0: K pair {4s,4s+1}; 1: {4s+2,4s+3}
    const int l15  = lane & 15;

    // hoist per-lane normalization constants (k set is fixed per lane)
    float mn[3][2], is[3][2];
#pragma unroll
    for (int s = 0; s < 3; ++s) {
        int k0 = 4 * s + 2 * half;
#pragma unroll
        for (int e = 0; e < 2; ++e) {
            int k = k0 + e;
            if (k < NF) { mn[s][e] = raw_mean[k]; is[s][e] = 1.0f / raw_std[k]; }
            else        { mn[s][e] = 0.0f;        is[s][e] = 1.0f; }
        }
    }

    for (int rt = 0; rt < TPW; ++rt) {
        const int rowTile = blockIdx.x * RT_PER_BLOCK + rt * WAVES + wave;
        const int row     = rowTile * 16 + l15;     // this lane's A-matrix row

        // prefetch next row-tile's x (global_prefetch_b8)
        if (rt + 1 < TPW) {
            const long nrow = (long)(rowTile + WAVES) * 16 + l15;
            __builtin_prefetch(x + nrow * NF, 0, 0);
        }

        // ---- A fragments: z_aug for 3 k-steps (ISA 16x4 f32 A layout) ----
        v2f afrag[3];
#pragma unroll
        for (int s = 0; s < 3; ++s) {
            int k0 = 4 * s + 2 * half;
            if (k0 >= NF) {                 // k = 10 (bias=1), 11 (pad=0)
                afrag[s].x = 1.0f;
                afrag[s].y = 0.0f;
            } else {
                v2f xv = *(const v2f*)(x + (long)row * NF + k0);
                float v0 = xv.x, v1 = xv.y;
                if (k0     == crp) v0 = __logf(fmaxf(v0 * 10.0f, 1e-6f));
                if (k0 + 1 == crp) v1 = __logf(fmaxf(v1 * 10.0f, 1e-6f));
                afrag[s].x = (v0 - mn[s][0]) * is[s][0];
                afrag[s].y = (v1 - mn[s][1]) * is[s][1];
            }
        }

        v8f acc = {0.f, 0.f, 0.f, 0.f, 0.f, 0.f, 0.f, 0.f};

#pragma unroll 2
        for (int t = 0; t < NTILES; ++t) {
            const float* col = &sW1T[(t * 16 + l15) * KP + 2 * half];
            v2f b0 = *(const v2f*)(col + 0);
            v2f b1 = *(const v2f*)(col + 4);
            v2f b2 = *(const v2f*)(col + 8);

            v8f c = {0.f, 0.f, 0.f, 0.f, 0.f, 0.f, 0.f, 0.f};
            c = __builtin_amdgcn_wmma_f32_16x16x4_f32(false, afrag[0], false, b0,
                                                      (short)0, c, false, false);
            c = __builtin_amdgcn_wmma_f32_16x16x4_f32(false, afrag[1], false, b1,
                                                      (short)0, c, false, false);
            c = __builtin_amdgcn_wmma_f32_16x16x4_f32(false, afrag[2], false, b2,
                                                      (short)0, c, false, false);

            const float w2v = sW2[t * 16 + l15];
#pragma unroll
            for (int i = 0; i < 8; ++i)
                acc[i] = fmaf(gelu_exact(c[i]), w2v, acc[i]);
        }

        // reduce over N within each 16-lane half (C layout: lanes0-15 -> M=i,
        // lanes16-31 -> M=i+8); masks < 16 keep the halves independent.
#pragma unroll
        for (int i = 0; i < 8; ++i) {
            float v = acc[i];
            v += __shfl_xor(v, 1, 32);
            v += __shfl_xor(v, 2, 32);
            v += __shfl_xor(v, 4, 32);
            v += __shfl_xor(v, 8, 32);
            acc[i] = v;
        }
        if (l15 == 0) {
            const int base = rowTile * 16 + half * 8;
#pragma unroll
            for (int i = 0; i < 8; ++i)
                out[base + i] = acc[i] + cbias;
        }
    }
}

extern "C" void kernel_launch(void* const* d_in, const int* in_sizes, int n_in,
                              void* d_out, int out_size, void* d_ws, size_t ws_size,
                              hipStream_t stream) {
    (void)in_sizes; (void)n_in; (void)out_size; (void)d_ws; (void)ws_size;
    const float* x        = (const float*)d_in[0];
    const float* raw_mean = (const float*)d_in[1];
    const float* raw_std  = (const float*)d_in[2];
    const float* w1f      = (const float*)d_in[3];
    const float* b1f      = (const float*)d_in[4];
    const float* w2f      = (const float*)d_in[5];
    const float* b2f      = (const float*)d_in[6];
    const float* w1p      = (const float*)d_in[7];
    const float* b1p      = (const float*)d_in[8];
    const float* w2p      = (const float*)d_in[9];
    const float* b2p      = (const float*)d_in[10];
    const int*   crp      = (const int*)d_in[11];
    float* out            = (float*)d_out;

    hipLaunchKernelGGL(nam_wmma_kernel, dim3(NBLOCKS), dim3(THREADS), 0, stream,
                       x, raw_mean, raw_std, w1f, b1f, w2f, b2f,
                       w1p, b1p, w2p, b2p, crp, out);
}